// ContextLearning_44229573214333
// MI455X (gfx1250) — compile-verified
//
#include <hip/hip_runtime.h>
#include <math.h>

#define CIN   64
#define NB    4          // batch
#define HH    128
#define WW    128
#define HWSZ  (HH*WW)    // 16384
#define KTOT  (CIN*9)    // 576

typedef __attribute__((ext_vector_type(2))) float v2f;
typedef __attribute__((ext_vector_type(8))) float v8f;

// ---------------------------------------------------------------------------
// Kernel 0: repack dw (4,64,64,3,3) -> wtp[branch][r/2][o][2], r = k*64 + c.
// Gives each WMMA lane a coalesced b64 load of B rows (r, r+1) for column o.
// ---------------------------------------------------------------------------
__global__ __launch_bounds__(256) void repack_kernel(
    const float* __restrict__ dw, float* __restrict__ wtp)
{
  const int idx = blockIdx.x * 256 + threadIdx.x;   // 0..147455
  const int branch = idx / (KTOT * CIN);
  const int rem    = idx - branch * KTOT * CIN;
  const int r      = rem >> 6;          // 0..575  (k*64 + c)
  const int n      = rem & 63;          // output channel
  const int k      = r >> 6;
  const int c      = r & 63;
  const float v = dw[(((size_t)(branch * CIN + n) * CIN + c) * 9) + k];
  wtp[(((size_t)branch * (KTOT / 2) + (r >> 1)) * CIN + n) * 2 + (r & 1)] = v;
}

// ---------------------------------------------------------------------------
// Kernel 1: offset convs (4 branches, 64->18 ch, 3x3 dilated) + bias + ReLU
// ---------------------------------------------------------------------------
__global__ __launch_bounds__(128) void offsets_kernel(
    const float* __restrict__ x, const float* __restrict__ ow,
    const float* __restrict__ ob, float* __restrict__ offbuf)
{
  __shared__ float wlds[18 * CIN * 9];   // 10368 floats = 41.5 KB
  const int j   = blockIdx.y;            // conv branch (dilation index)
  const int tid = threadIdx.x;
  const float* wj = ow + j * 18 * CIN * 9;
  for (int i = tid; i < 18 * CIN * 9; i += 128) wlds[i] = wj[i];
  __syncthreads();

  static const int rates[4] = {2, 3, 5, 9};
  const int d = rates[j];

  const int p  = blockIdx.x * 128 + tid;   // 0..65535 = b*HW + hw
  const int b  = p >> 14;
  const int hw = p & (HWSZ - 1);
  const int h  = hw >> 7;
  const int w  = hw & (WW - 1);

  float acc[18];
#pragma unroll
  for (int o = 0; o < 18; ++o) acc[o] = 0.f;

  const float* xb = x + b * CIN * HWSZ;
  // branchless: clamped address, mask multiplied into the tap value
  int   tidx[9];
  float tmask[9];
#pragma unroll
  for (int t = 0; t < 9; ++t) {
    const int yy = h + (t / 3 - 1) * d;
    const int xx = w + (t % 3 - 1) * d;
    const bool ok = (yy >= 0) && (yy < HH) && (xx >= 0) && (xx < WW);
    const int yc = min(max(yy, 0), HH - 1);
    const int xc = min(max(xx, 0), WW - 1);
    tidx[t]  = yc * WW + xc;
    tmask[t] = ok ? 1.f : 0.f;
  }
  for (int c = 0; c < CIN; ++c) {
    const float* plane = xb + c * HWSZ;
#pragma unroll
    for (int t = 0; t < 9; ++t) {
      const float v = plane[tidx[t]] * tmask[t];
#pragma unroll
      for (int o = 0; o < 18; ++o)
        acc[o] = fmaf(v, wlds[(o * CIN + c) * 9 + t], acc[o]);
    }
  }
  float* dst = offbuf + ((size_t)(j * NB + b) * 18) * HWSZ + hw;
#pragma unroll
  for (int o = 0; o < 18; ++o) {
    float r = acc[o] + ob[j * 18 + o];
    dst[(size_t)o * HWSZ] = r > 0.f ? r : 0.f;
  }
}

// ---------------------------------------------------------------------------
// Kernel 2: deformable conv as WMMA GEMM, reduction index r = k*64 + c.
// One wave = 16 consecutive pixels x 64 output channels (4 N-tiles, v8f each).
// Per (pixel,tap): 4 clamped indices + 4 masked corner weights precomputed in
// LDS; hoisted into registers once per k-block (16 WMMA steps).
// ---------------------------------------------------------------------------
__global__ __launch_bounds__(128) void deform_kernel(
    const float* __restrict__ x, const float* __restrict__ wtp,
    const float* __restrict__ offbuf, float* __restrict__ catbuf)
{
  __shared__ int   s_ix[4][144][4];
  __shared__ float s_cw[4][144][4];

  const int tid  = threadIdx.x;
  const int wave = tid >> 5;
  const int lane = tid & 31;
  const int m    = lane & 15;    // pixel within tile / column within N-tile
  const int half = lane >> 4;

  const int branch = blockIdx.y;
  static const int rates[4] = {2, 3, 5, 9};
  const int rate = rates[branch];
  const int j    = (branch + 3) & 3;   // offsets come from conv (i-1)%4

  const int tileIdx = blockIdx.x * 4 + wave;   // 4096 tiles per branch
  const int b   = tileIdx >> 10;
  const int pos = (tileIdx & 1023) << 4;       // base pixel (16 per tile, same row)
  const int h   = pos >> 7;
  const int w   = (pos & 127) + m;

  // ---- per-(pixel,tap) clamped indices + masked weights into LDS ----
  const float* offb = offbuf + ((size_t)(j * NB + b) * 18) * HWSZ + h * WW + w;
  for (int k = half; k < 9; k += 2) {
    const float dy = offb[(size_t)(2 * k + 0) * HWSZ];
    const float dx = offb[(size_t)(2 * k + 1) * HWSZ];
    const float py = (float)h + (float)((k / 3 - 1) * rate) + dy;
    const float px = (float)w + (float)((k % 3 - 1) * rate) + dx;
    const float y0f = floorf(py), x0f = floorf(px);
    const int y0 = (int)y0f, x0 = (int)x0f;
    const float fy = py - y0f, fx = px - x0f;
    const float wy0 = 1.f - fy, wx0 = 1.f - fx;
    const bool yv0 = (y0 >= 0) && (y0 < HH);
    const bool yv1 = (y0 >= -1) && (y0 < HH - 1);
    const bool xv0 = (x0 >= 0) && (x0 < WW);
    const bool xv1 = (x0 >= -1) && (x0 < WW - 1);
    const int yc0 = min(max(y0, 0), HH - 1);
    const int yc1 = min(max(y0 + 1, 0), HH - 1);
    const int xc0 = min(max(x0, 0), WW - 1);
    const int xc1 = min(max(x0 + 1, 0), WW - 1);
    const int e = m * 9 + k;
    s_ix[wave][e][0] = yc0 * WW + xc0;
    s_ix[wave][e][1] = yc0 * WW + xc1;
    s_ix[wave][e][2] = yc1 * WW + xc0;
    s_ix[wave][e][3] = yc1 * WW + xc1;
    s_cw[wave][e][0] = (yv0 && xv0) ? wy0 * wx0 : 0.f;
    s_cw[wave][e][1] = (yv0 && xv1) ? wy0 * fx  : 0.f;
    s_cw[wave][e][2] = (yv1 && xv0) ? fy  * wx0 : 0.f;
    s_cw[wave][e][3] = (yv1 && xv1) ? fy  * fx  : 0.f;
  }
  __syncthreads();   // lane-halves exchange tap parities

  const float* xb = x + (size_t)b * CIN * HWSZ;
  const float* wbb = wtp + (size_t)branch * (KTOT / 2) * CIN * 2;

  v8f acc[4] = {};   // 4 N-tiles of 16 output channels

  for (int k = 0; k < 9; ++k) {
    const int e = m * 9 + k;
    const int4   ix = *(const int4*)  (&s_ix[wave][e][0]);
    const float4 cw = *(const float4*)(&s_cw[wave][e][0]);
#pragma unroll 2
    for (int cc = 0; cc < CIN; cc += 4) {
      const int c0 = cc + 2 * half;          // channels c0, c0+1 for this lane
      const float* p0 = xb + (size_t)c0 * HWSZ;
      const float* p1 = p0 + HWSZ;
      float v0 = p0[ix.x] * cw.x;
      v0 = fmaf(p0[ix.y], cw.y, v0);
      v0 = fmaf(p0[ix.z], cw.z, v0);
      v0 = fmaf(p0[ix.w], cw.w, v0);
      float v1 = p1[ix.x] * cw.x;
      v1 = fmaf(p1[ix.y], cw.y, v1);
      v1 = fmaf(p1[ix.z], cw.z, v1);
      v1 = fmaf(p1[ix.w], cw.w, v1);
      v2f A;
      A.x = v0;
      A.y = v1;
      const int rp = (k * CIN + c0) >> 1;    // packed row-pair index (r0 even)
      const float* bp = wbb + ((size_t)rp * CIN + m) * 2;
#pragma unroll
      for (int nt = 0; nt < 4; ++nt) {
        const v2f Bf = *(const v2f*)(bp + nt * 32);   // rows r0, r0+1, col nt*16+m
        acc[nt] = __builtin_amdgcn_wmma_f32_16x16x4_f32(
            false, A, false, Bf, (short)0, acc[nt], false, false);
      }
    }
  }

  // D layout: component v -> pixel M = v + 8*half, channel N = nt*16 + m
#pragma unroll
  for (int nt = 0; nt < 4; ++nt) {
    const int o = nt * 16 + m;
    float* dst = catbuf + ((size_t)(branch * NB + b) * CIN + o) * HWSZ
               + pos + 8 * half;
    float4 lo = make_float4(acc[nt][0], acc[nt][1], acc[nt][2], acc[nt][3]);
    float4 hi = make_float4(acc[nt][4], acc[nt][5], acc[nt][6], acc[nt][7]);
    *(float4*)(dst)     = lo;
    *(float4*)(dst + 4) = hi;
  }
}

// ---------------------------------------------------------------------------
// Kernel 3: global average pool over each (b, channel) plane of cat
// ---------------------------------------------------------------------------
__global__ __launch_bounds__(256) void mean_kernel(
    const float* __restrict__ catbuf, float* __restrict__ g)
{
  __shared__ float red[256];
  const int q  = blockIdx.x;        // b*256 + ch
  const int b  = q >> 8;
  const int ch = q & 255;
  const int i  = ch >> 6;
  const int o  = ch & 63;
  const float* plane = catbuf + ((size_t)(i * NB + b) * CIN + o) * HWSZ;
  float s = 0.f;
  for (int t = threadIdx.x; t < HWSZ; t += 256) s += plane[t];
  red[threadIdx.x] = s;
  __syncthreads();
  for (int st = 128; st > 0; st >>= 1) {
    if (threadIdx.x < st) red[threadIdx.x] += red[threadIdx.x + st];
    __syncthreads();
  }
  if (threadIdx.x == 0) g[q] = red[0] * (1.f / 16384.f);
}

// ---------------------------------------------------------------------------
// Kernel 4: SE MLP -> scales = 1 + sigmoid(relu(g@W1^T+b1)@W2^T+b2)
// ---------------------------------------------------------------------------
__global__ __launch_bounds__(256) void se_kernel(
    const float* __restrict__ g,
    const float* __restrict__ w1, const float* __restrict__ b1,
    const float* __restrict__ w2, const float* __restrict__ b2,
    float* __restrict__ scales)
{
  __shared__ float gl[1024];
  __shared__ float h1[256];   // [b*64+u]
  const int tid = threadIdx.x;
  for (int i = tid; i < 1024; i += 256) gl[i] = g[i];
  __syncthreads();
  {
    const int b = tid >> 6, u = tid & 63;
    float s = b1[u];
    for (int ch = 0; ch < 256; ++ch)
      s = fmaf(gl[b * 256 + ch], w1[u * 256 + ch], s);
    h1[tid] = s > 0.f ? s : 0.f;
  }
  __syncthreads();
#pragma unroll
  for (int q = 0; q < 4; ++q) {
    const int idx = q * 256 + tid;        // 0..1023
    const int b = idx >> 8, ch = idx & 255;
    float s = b2[ch];
    for (int u = 0; u < 64; ++u)
      s = fmaf(h1[b * 64 + u], w2[ch * 64 + u], s);
    scales[idx] = 1.f + 1.f / (1.f + expf(-s));
  }
}

// ---------------------------------------------------------------------------
// Kernel 5: out = x + sum_i scales[b, i*64+c] * cat[i][b][c]
// ---------------------------------------------------------------------------
__global__ __launch_bounds__(256) void fuse_kernel(
    const float* __restrict__ x, const float* __restrict__ catbuf,
    const float* __restrict__ scales, float* __restrict__ out)
{
  const int idx = blockIdx.x * 256 + threadIdx.x;   // 0..4194303
  const int b   = idx >> 20;
  const int rem = idx & ((1 << 20) - 1);
  const int c   = rem >> 14;
  const int p   = rem & (HWSZ - 1);
  float v = x[idx];
#pragma unroll
  for (int i = 0; i < 4; ++i) {
    const float sc = scales[b * 256 + i * 64 + c];
    v = fmaf(sc, catbuf[((size_t)(i * NB + b) * CIN + c) * HWSZ + p], v);
  }
  out[idx] = v;
}

// ---------------------------------------------------------------------------
extern "C" void kernel_launch(void* const* d_in, const int* in_sizes, int n_in,
                              void* d_out, int out_size, void* d_ws, size_t ws_size,
                              hipStream_t stream) {
  (void)in_sizes; (void)n_in; (void)out_size; (void)ws_size;
  const float* x  = (const float*)d_in[0];
  const float* dw = (const float*)d_in[1];
  const float* ow = (const float*)d_in[2];
  const float* ob = (const float*)d_in[3];
  const float* w1 = (const float*)d_in[4];
  const float* b1 = (const float*)d_in[5];
  const float* w2 = (const float*)d_in[6];
  const float* b2 = (const float*)d_in[7];

  float* ws     = (float*)d_ws;
  float* offbuf = ws;                                       //  4*4*18*16384
  float* catbuf = offbuf + (size_t)4 * NB * 18 * HWSZ;      //  4*4*64*16384
  float* gbuf   = catbuf + (size_t)4 * NB * CIN * HWSZ;     //  1024
  float* sbuf   = gbuf + 1024;                              //  1024
  float* wtpbuf = sbuf + 1024;                              //  4*576*64
  float* out    = (float*)d_out;

  repack_kernel <<<576, 256, 0, stream>>>(dw, wtpbuf);
  offsets_kernel<<<dim3(512, 4), 128, 0, stream>>>(x, ow, ob, offbuf);
  deform_kernel <<<dim3(1024, 4), 128, 0, stream>>>(x, wtpbuf, offbuf, catbuf);
  mean_kernel   <<<1024, 256, 0, stream>>>(catbuf, gbuf);
  se_kernel     <<<1, 256, 0, stream>>>(gbuf, w1, b1, w2, b2, sbuf);
  fuse_kernel   <<<16384, 256, 0, stream>>>(x, catbuf, sbuf, out);
}